// LNO1d_46626164966080
// MI455X (gfx1250) — compile-verified
//
#include <hip/hip_runtime.h>
#include <math.h>

#define NB     64
#define NS     8192
#define NW     4
#define NMODES 32
#define NHID   256
#define TWO_PI 6.28318530717958647692f

typedef __attribute__((ext_vector_type(2))) float v2f;
typedef __attribute__((ext_vector_type(8))) float v8f;

__device__ __forceinline__ float2 cadd(float2 a, float2 b){ return make_float2(a.x+b.x, a.y+b.y); }
__device__ __forceinline__ float2 csub(float2 a, float2 b){ return make_float2(a.x-b.x, a.y-b.y); }
__device__ __forceinline__ float2 cmul(float2 a, float2 b){ return make_float2(a.x*b.x - a.y*b.y, a.x*b.y + a.y*b.x); }

// ---------------------------------------------------------------- lift: h[b,w,s] = sin(fc0([x,grid]))
__global__ __launch_bounds__(256) void lift_kernel(const float* __restrict__ x,
                                                   const float* __restrict__ fc0_w,
                                                   const float* __restrict__ fc0_b,
                                                   float* __restrict__ h) {
  int n = blockIdx.x * 256 + threadIdx.x;        // n = b*NS + z, in [0, NB*NS)
  int b = n >> 13;                                // NS = 2^13
  int z = n & (NS - 1);
  float xv = x[n];
  float g  = (float)z * (1.0f / (float)(NS - 1));
  #pragma unroll
  for (int w = 0; w < NW; ++w) {
    float hv = __sinf(xv * fc0_w[w] + g * fc0_w[NW + w] + fc0_b[w]);
    h[((size_t)(b * NW + w)) * NS + z] = hv;
  }
}

// ---------------------------------------------------------------- Stockham autosort radix-2 FFT in LDS
__device__ float2* stockham(float2* bufA, float2* bufB, float sign) {
  float2* src = bufA;
  float2* dst = bufB;
  int n = NS;
  int ls = 0;                                     // s = 1 << ls
  while (n > 1) {
    int m = n >> 1;
    float theta = sign * (TWO_PI / (float)n);
    __syncthreads();
    int s = 1 << ls;
    for (int idx = threadIdx.x; idx < (NS >> 1); idx += 256) {
      int p = idx >> ls;
      int q = idx & (s - 1);
      float2 u = src[q + s * p];
      float2 v = src[q + s * (p + m)];
      float sn, cs;
      __sincosf(theta * (float)p, &sn, &cs);
      float2 w = make_float2(cs, sn);             // exp(sign * i*2*pi*p/n)
      dst[q + s * (2 * p)]     = cadd(u, v);
      dst[q + s * (2 * p + 1)] = cmul(csub(u, v), w);
    }
    float2* t = src; src = dst; dst = t;
    n = m; ++ls;
  }
  __syncthreads();
  return src;
}

// forward FFT of real rows h[r][*] -> alpha[r][*]
__global__ __launch_bounds__(256) void fft_fwd_kernel(const float* __restrict__ h,
                                                      float2* __restrict__ alpha) {
  extern __shared__ float2 lds[];
  float2* A = lds;
  float2* Bb = lds + NS;
  int r = blockIdx.x;                             // r = b*NW + i  (256 rows)
  for (int i = threadIdx.x; i < NS; i += 256)
    A[i] = make_float2(h[(size_t)r * NS + i], 0.0f);
  float2* res = stockham(A, Bb, -1.0f);
  for (int i = threadIdx.x; i < NS; i += 256)
    alpha[(size_t)r * NS + i] = res[i];
}

// inverse FFT of res1 rows -> real part into y[b][z][o]  (initializes y)
__global__ __launch_bounds__(256) void fft_inv_kernel(const float2* __restrict__ res1,
                                                      float* __restrict__ y) {
  extern __shared__ float2 lds[];
  float2* A = lds;
  float2* Bb = lds + NS;
  int r = blockIdx.x;                             // r = b*NW + o
  int b = r >> 2, o = r & 3;
  for (int i = threadIdx.x; i < NS; i += 256)
    A[i] = res1[(size_t)r * NS + i];
  float2* res = stockham(A, Bb, +1.0f);
  const float inv = 1.0f / (float)NS;
  for (int z = threadIdx.x; z < NS; z += 256)
    y[((size_t)b * NS + z) * NW + o] = res[z].x * inv;
}

// ---------------------------------------------------------------- Hw[ok][i][x] = residue/(j*w_x - pole)
// Layout transposed to [o*32+k][i][x] so the K-reductions stream coalesced over x.
__global__ __launch_bounds__(256) void hw_kernel(const float2* __restrict__ pole,
                                                 const float2* __restrict__ residue,
                                                 float2* __restrict__ Hw) {
  size_t n = (size_t)blockIdx.x * 256 + threadIdx.x;   // ((ok*4 + i)*NS + x), S*512 entries
  int x = (int)(n & (NS - 1));
  int t = (int)(n >> 13);
  int i  = t & 3;
  int ok = t >> 2;                                // o*32 + k
  int o = ok >> 5, k = ok & 31;
  float f = (x < NS / 2) ? (float)x : (float)(x - NS);
  float omega = TWO_PI * f * ((float)(NS - 1) / (float)NS);   // 2*pi*fftfreq(S, dt=1/(S-1))
  int pidx = (i * NW + o) * NMODES + k;
  float2 p = pole[pidx];
  float2 rr = residue[pidx];
  float dr = -p.x, di = omega - p.y;              // d = j*omega - pole
  float inv = 1.0f / (dr * dr + di * di);
  Hw[n] = make_float2((rr.x * dr + rr.y * di) * inv,
                      (rr.y * dr - rr.x * di) * inv);
}

// G[i*4+o][x] = sum_k Hw[(o*32+k)][i][x]
__global__ __launch_bounds__(256) void g_kernel(const float2* __restrict__ Hw,
                                                float2* __restrict__ G) {
  int n = blockIdx.x * 256 + threadIdx.x;         // io*NS + x  (16*S entries)
  int x  = n & (NS - 1);
  int io = n >> 13;                               // i*4 + o
  int i = io >> 2, o = io & 3;
  float2 acc = make_float2(0.f, 0.f);
  #pragma unroll
  for (int k = 0; k < NMODES; ++k)
    acc = cadd(acc, Hw[((size_t)((o * NMODES + k) * NW + i)) * NS + x]);
  G[n] = acc;
}

// res1[b][o][x] = sum_i alpha[b][i][x] * G[i*4+o][x]
__global__ __launch_bounds__(256) void res1_kernel(const float2* __restrict__ alpha,
                                                   const float2* __restrict__ G,
                                                   float2* __restrict__ res1) {
  int n = blockIdx.x * 256 + threadIdx.x;         // NB*NS
  int b = n >> 13;
  int x = n & (NS - 1);
  float2 a[NW];
  #pragma unroll
  for (int i = 0; i < NW; ++i) a[i] = alpha[(size_t)(b * NW + i) * NS + x];
  #pragma unroll
  for (int o = 0; o < NW; ++o) {
    float2 acc = make_float2(0.f, 0.f);
    #pragma unroll
    for (int i = 0; i < NW; ++i)
      acc = cadd(acc, cmul(a[i], G[((size_t)(i * NW + o)) * NS + x]));
    res1[(size_t)(b * NW + o) * NS + x] = acc;
  }
}

// res2[b][o*32+k] = -sum_{i,x} alpha[b][i][x] * Hw[(o*32+k)][i][x]   (coalesced over x)
__global__ __launch_bounds__(256) void res2_kernel(const float2* __restrict__ alpha,
                                                   const float2* __restrict__ Hw,
                                                   float2* __restrict__ res2) {
  int blk = blockIdx.x;                           // b*128 + ok
  int b = blk >> 7;
  int ok = blk & 127;
  float2 acc = make_float2(0.f, 0.f);
  for (int i = 0; i < NW; ++i) {
    const float2* ar = alpha + (size_t)(b * NW + i) * NS;
    const float2* hw = Hw + (size_t)(ok * NW + i) * NS;
    for (int x = threadIdx.x; x < NS; x += 256)
      acc = cadd(acc, cmul(ar[x], hw[x]));
  }
  __shared__ float2 red[256];
  red[threadIdx.x] = acc;
  __syncthreads();
  for (int st = 128; st >= 1; st >>= 1) {
    if (threadIdx.x < st)
      red[threadIdx.x] = cadd(red[threadIdx.x], red[threadIdx.x + st]);
    __syncthreads();
  }
  if (threadIdx.x == 0) res2[blk] = make_float2(-red[0].x, -red[0].y);
}

// y[b][z][o] += Re(sum_{i,k} res2[b][i*32+k] * exp(pole[i][o][k]*t_z))/S + skip
__global__ __launch_bounds__(256) void x2skip_kernel(const float2* __restrict__ res2,
                                                     const float2* __restrict__ pole,
                                                     const float* __restrict__ w0_w,
                                                     const float* __restrict__ w0_b,
                                                     const float* __restrict__ h,
                                                     float* __restrict__ y) {
  __shared__ float2 Els[512];                     // [o][i*32+k]
  int z = blockIdx.x;
  float tz = (float)z * (1.0f / (float)(NS - 1));
  for (int e = threadIdx.x; e < 512; e += 256) {
    int o = e >> 7;
    int ik = e & 127;
    int i = ik >> 5, k = ik & 31;
    float2 p = pole[(i * NW + o) * NMODES + k];
    float mag = __expf(p.x * tz);
    float sn, cs;
    __sincosf(p.y * tz, &sn, &cs);
    Els[e] = make_float2(mag * cs, mag * sn);
  }
  __syncthreads();
  int b = threadIdx.x >> 2;                       // 64 b * 4 o = 256 threads
  int o = threadIdx.x & 3;
  const float2* rb = res2 + b * 128;
  const float2* eb = Els + o * 128;
  float accr = 0.f;
  #pragma unroll 4
  for (int j = 0; j < 128; ++j) {
    float2 rv = rb[j];
    float2 ev = eb[j];
    accr += rv.x * ev.x - rv.y * ev.y;            // real part only
  }
  float x2v = accr * (1.0f / (float)NS);
  float sk = w0_b[o];
  #pragma unroll
  for (int i = 0; i < NW; ++i)
    sk += w0_w[o * NW + i] * h[(size_t)(b * NW + i) * NS + z];
  y[((size_t)b * NS + z) * NW + o] += x2v + sk;
}

// ---------------------------------------------------------------- MLP head with WMMA f32 16x16x4
// out[r] = sin(y[r,:4] @ fc1_w + fc1_b) @ fc2_w + fc2_b ; one wave handles 16 rows.
__global__ __launch_bounds__(256) void fc_kernel(const float* __restrict__ y,
                                                 const float* __restrict__ fc1_w,
                                                 const float* __restrict__ fc1_b,
                                                 const float* __restrict__ fc2_w,
                                                 const float* __restrict__ fc2_b,
                                                 float* __restrict__ out) {
  int lane  = threadIdx.x & 31;
  int wave  = threadIdx.x >> 5;
  int rowBase = (blockIdx.x * 8 + wave) * 16;
  int mlane = lane & 15;
  int kbase = (lane >> 4) * 2;                    // lanes 0-15: K=0,1 ; lanes 16-31: K=2,3

  int r = rowBase + mlane;
  v2f a;
  a.x = y[(size_t)r * NW + kbase];
  a.y = y[(size_t)r * NW + kbase + 1];

  float acc[8];
  #pragma unroll
  for (int v = 0; v < 8; ++v) acc[v] = 0.f;

  for (int nt = 0; nt < NHID / 16; ++nt) {
    int n = nt * 16 + mlane;
    v2f bb;
    bb.x = fc1_w[(size_t)kbase * NHID + n];
    bb.y = fc1_w[(size_t)(kbase + 1) * NHID + n];
    float bias = fc1_b[n];
    v8f c;
    #pragma unroll
    for (int v = 0; v < 8; ++v) c[v] = bias;      // fold fc1 bias into accumulator
    c = __builtin_amdgcn_wmma_f32_16x16x4_f32(false, a, false, bb, (short)0, c, false, false);
    float w2 = fc2_w[n];
    #pragma unroll
    for (int v = 0; v < 8; ++v) acc[v] += __sinf(c[v]) * w2;
  }
  // reduce the 16 N-columns held across lanes of each half-wave
  #pragma unroll
  for (int m = 8; m >= 1; m >>= 1) {
    #pragma unroll
    for (int v = 0; v < 8; ++v) acc[v] += __shfl_xor(acc[v], m, 16);
  }
  float b2 = fc2_b[0];
  if (mlane == 0) {
    int roff = rowBase + (lane >> 4) * 8;         // lanes 0-15: rows 0-7 ; 16-31: rows 8-15
    #pragma unroll
    for (int v = 0; v < 8; ++v) out[roff + v] = acc[v] + b2;
  }
}

// ----------------------------------------------------------------
extern "C" void kernel_launch(void* const* d_in, const int* in_sizes, int n_in,
                              void* d_out, int out_size, void* d_ws, size_t ws_size,
                              hipStream_t stream) {
  const float*  x     = (const float*)d_in[0];
  const float*  fc0_w = (const float*)d_in[1];
  const float*  fc0_b = (const float*)d_in[2];
  const float2* pole  = (const float2*)d_in[3];
  const float2* resi  = (const float2*)d_in[4];
  const float*  w0_w  = (const float*)d_in[5];
  const float*  w0_b  = (const float*)d_in[6];
  const float*  fc1_w = (const float*)d_in[7];
  const float*  fc1_b = (const float*)d_in[8];
  const float*  fc2_w = (const float*)d_in[9];
  const float*  fc2_b = (const float*)d_in[10];
  float* out = (float*)d_out;

  char* ws = (char*)d_ws;
  float*  h     = (float*) ws;  ws += (size_t)NB * NW * NS * sizeof(float);    // 8 MB
  float2* alpha = (float2*)ws;  ws += (size_t)NB * NW * NS * sizeof(float2);   // 16 MB
  float2* Hw    = (float2*)ws;  ws += (size_t)NS * 512 * sizeof(float2);       // 33.5 MB
  float2* G     = (float2*)ws;  ws += (size_t)NS * 16 * sizeof(float2);        // 1 MB
  float2* res1  = (float2*)ws;  ws += (size_t)NB * NW * NS * sizeof(float2);   // 16 MB
  float2* res2  = (float2*)ws;  ws += (size_t)NB * 128 * sizeof(float2);       // 64 KB
  float*  yb    = (float*) ws;  ws += (size_t)NB * NS * NW * sizeof(float);    // 8 MB

  const size_t fftLds = 2 * NS * sizeof(float2);  // 128 KB Stockham ping-pong

  lift_kernel <<<(NB * NS) / 256, 256, 0, stream>>>(x, fc0_w, fc0_b, h);
  fft_fwd_kernel<<<NB * NW, 256, fftLds, stream>>>(h, alpha);
  hw_kernel   <<<(NS * 512) / 256, 256, 0, stream>>>(pole, resi, Hw);
  g_kernel    <<<(NS * 16) / 256, 256, 0, stream>>>(Hw, G);
  res1_kernel <<<(NB * NS) / 256, 256, 0, stream>>>(alpha, G, res1);
  res2_kernel <<<NB * 128, 256, 0, stream>>>(alpha, Hw, res2);
  fft_inv_kernel<<<NB * NW, 256, fftLds, stream>>>(res1, yb);
  x2skip_kernel<<<NS, 256, 0, stream>>>(res2, pole, w0_w, w0_b, h, yb);
  fc_kernel   <<<(NB * NS) / 128, 256, 0, stream>>>(yb, fc1_w, fc1_b, fc2_w, fc2_b, out);
}